// CausalAttention_13159779795568
// MI455X (gfx1250) — compile-verified
//
#include <hip/hip_runtime.h>

// ---------------------------------------------------------------------------
// Single-query attention, fused flash-style (read outputs [16,4096,2048] f32
// from HBM exactly once):
//   logits[b,n] = <last_h[b], outputs[b,n]> / sqrt(H)
//   attn = softmax_n(logits);  ctx[b] = sum_n attn[b,n] * outputs[b,n]
// d_out = [ ctx (B*H) | attn (B*N) ]  (tuple return order)
//
// Kernel 1: grid (CHUNKS, B). Each wave streams its rows once (b128 NT loads),
//           computes the logit via VALU dot + wave32 shfl_xor reduce, and does
//           the online-softmax ctx accumulation from the SAME registers.
//           Writes per-chunk (m, s, partial_ctx[H]) to d_ws and raw logits
//           into the attn region of d_out.
// Kernel 2: per batch, combines chunk partials. The ctx combine
//           ctx = (softmax chunk weights)[64] x P[64,2048] is mapped onto
//           v_wmma_f32_16x16x4_f32 (A rows = broadcast weights, B = 4x16 slab
//           of P, D = 16-wide h tile), run after attn is normalized in place.
// ---------------------------------------------------------------------------

typedef __attribute__((ext_vector_type(2))) float v2f;
typedef __attribute__((ext_vector_type(4))) float v4f;
typedef __attribute__((ext_vector_type(8))) float v8f;

constexpr int B = 16, N = 4096, H = 2048;
constexpr int CHUNKS = 64;                 // chunks per batch row-dim
constexpr int ROWS   = N / CHUNKS;         // 64 rows per chunk
constexpr int WAVES  = 8, TPB = WAVES * 32;
constexpr int RPW    = ROWS / WAVES;       // 8 rows per wave
constexpr int VPT    = H / (32 * 4);       // 16 float4 per lane covers one row
constexpr float SCALE = 0.022097086912079608f; // 1/sqrt(2048)

__global__ __launch_bounds__(TPB)
void flash_partial(const float* __restrict__ outputs,
                   const float* __restrict__ last_h,
                   float* __restrict__ logits_raw,   // aliases attn region of d_out
                   float* __restrict__ pctx,         // [B*CHUNKS*H]
                   float* __restrict__ pm,           // [B*CHUNKS]
                   float* __restrict__ ps)           // [B*CHUNKS]
{
    const int chunk = blockIdx.x;
    const int b     = blockIdx.y;
    const int tid   = threadIdx.x;
    const int lane  = tid & 31;
    const int wave  = tid >> 5;

    __shared__ __align__(16) float qsh[H];     // last_h[b]
    __shared__ __align__(16) float ctxbuf[H];  // block partial ctx
    __shared__ float red_m[WAVES], red_s[WAVES];

    for (int i = tid; i < H; i += TPB) {
        qsh[i]    = last_h[(size_t)b * H + i];
        ctxbuf[i] = 0.0f;
    }
    __syncthreads();

    const v4f* q4 = (const v4f*)qsh;

    float m_w = -INFINITY, s_w = 0.0f;
    v4f acc[VPT];
#pragma unroll
    for (int i = 0; i < VPT; ++i) acc[i] = v4f{0.f, 0.f, 0.f, 0.f};

    const int nbase = chunk * ROWS + wave * RPW;
    for (int r = 0; r < RPW; ++r) {
        const int n = nbase + r;
        const v4f* rp = (const v4f*)(outputs + ((size_t)b * N + n) * H);

        // Stream the whole row into VGPRs (16 x b128 per lane, coalesced, NT).
        v4f rv[VPT];
#pragma unroll
        for (int i = 0; i < VPT; ++i)
            rv[i] = __builtin_nontemporal_load(rp + lane + 32 * i);

        // Per-lane partial dot, then wave32 all-reduce.
        float t = 0.0f;
#pragma unroll
        for (int i = 0; i < VPT; ++i) {
            v4f qv = q4[lane + 32 * i];
            t += rv[i].x * qv.x + rv[i].y * qv.y + rv[i].z * qv.z + rv[i].w * qv.w;
        }
#pragma unroll
        for (int off = 16; off > 0; off >>= 1)
            t += __shfl_xor(t, off, 32);

        const float logit = t * SCALE;
        if (lane == 0) logits_raw[(size_t)b * N + n] = logit;

        // Online softmax update; ctx accumulated from the registers we already
        // hold -> no second HBM pass over outputs.
        const float m_new = fmaxf(m_w, logit);
        const float resc  = __expf(m_w - m_new);
        const float p     = __expf(logit - m_new);
        s_w = s_w * resc + p;
#pragma unroll
        for (int i = 0; i < VPT; ++i)
            acc[i] = acc[i] * resc + rv[i] * p;
        m_w = m_new;
    }

    // Combine the 8 waves of this block (deterministic, turn-based).
    if (lane == 0) red_m[wave] = m_w;
    __syncthreads();
    float m_B = red_m[0];
#pragma unroll
    for (int w = 1; w < WAVES; ++w) m_B = fmaxf(m_B, red_m[w]);

    const float wgt = __expf(m_w - m_B);
    if (lane == 0) red_s[wave] = s_w * wgt;

    v4f* c4 = (v4f*)ctxbuf;
    for (int w = 0; w < WAVES; ++w) {
        if (wave == w) {
#pragma unroll
            for (int i = 0; i < VPT; ++i)
                c4[lane + 32 * i] += acc[i] * wgt;
        }
        __syncthreads();
    }

    float s_B = 0.0f;
#pragma unroll
    for (int w = 0; w < WAVES; ++w) s_B += red_s[w];

    float* out = pctx + ((size_t)b * CHUNKS + chunk) * H;
    for (int i = tid; i < H; i += TPB) out[i] = ctxbuf[i];
    if (tid == 0) {
        pm[b * CHUNKS + chunk] = m_B;
        ps[b * CHUNKS + chunk] = s_B;
    }
}

__global__ __launch_bounds__(256)
void combine(const float* __restrict__ pctx,
             const float* __restrict__ pm,
             const float* __restrict__ ps,
             float* __restrict__ ctx,
             float* attn)   // holds raw logits on entry; normalized in place
{
    const int b    = blockIdx.x;
    const int tid  = threadIdx.x;
    const int lane = tid & 31;
    const int wave = tid >> 5;

    __shared__ float wsh[CHUNKS];   // per-chunk softmax weight exp(m_c-gm)/gs
    __shared__ float sh_gm, sh_gs;

    if (tid == 0) {
        float gm = -INFINITY;
        for (int c = 0; c < CHUNKS; ++c) gm = fmaxf(gm, pm[b * CHUNKS + c]);
        float gs = 0.0f;
        for (int c = 0; c < CHUNKS; ++c) gs += ps[b * CHUNKS + c] * __expf(pm[b * CHUNKS + c] - gm);
        const float inv = 1.0f / gs;
        for (int c = 0; c < CHUNKS; ++c) wsh[c] = __expf(pm[b * CHUNKS + c] - gm) * inv;
        sh_gm = gm; sh_gs = gs;
    }
    __syncthreads();
    const float gm    = sh_gm;
    const float invgs = 1.0f / sh_gs;

    // Normalize attn in place (exact multiples of 256 -> no divergence).
    for (int i = tid; i < N; i += 256) {
        const size_t idx = (size_t)b * N + i;
        attn[idx] = __expf(attn[idx] - gm) * invgs;
    }

    // ctx[b, :] = w[64] x P[64, H] via v_wmma_f32_16x16x4_f32.
    // A (16x4 f32 layout): lanes 0-15 hold {K0,K1}, lanes 16-31 hold {K2,K3};
    // all 16 M rows identical (broadcast weights) -> every D row is the tile.
    // B (4x16 layout): VGPR0 = rows K0/K2 striped over lane halves, VGPR1 = K1/K3.
    const float* P    = pctx + (size_t)b * CHUNKS * H;
    const int nsel    = lane & 15;
    const int khalf   = (lane < 16) ? 0 : 2;

    for (int tile = wave; tile < H / 16; tile += WAVES) {   // 16 tiles per wave
        const int h0 = tile * 16;
        v8f d = {0.f, 0.f, 0.f, 0.f, 0.f, 0.f, 0.f, 0.f};
        for (int c0 = 0; c0 < CHUNKS; c0 += 4) {
            v2f a;  a.x = wsh[c0 + khalf];  a.y = wsh[c0 + khalf + 1];
            const float* col = P + (size_t)(c0 + khalf) * H + h0 + nsel;
            v2f bb; bb.x = col[0];          bb.y = col[H];
            d = __builtin_amdgcn_wmma_f32_16x16x4_f32(
                    /*neg_a=*/false, a, /*neg_b=*/false, bb,
                    /*c_mod=*/(short)0, d, /*reuse_a=*/false, /*reuse_b=*/false);
        }
        if (lane < 16) ctx[(size_t)b * H + h0 + lane] = d[0];
    }
}

extern "C" void kernel_launch(void* const* d_in, const int* in_sizes, int n_in,
                              void* d_out, int out_size, void* d_ws, size_t ws_size,
                              hipStream_t stream) {
    const float* outputs = (const float*)d_in[0];   // [B,N,H]
    const float* last_h  = (const float*)d_in[1];   // [B,H]

    float* ctx  = (float*)d_out;                       // B*H
    float* attn = (float*)d_out + (size_t)B * H;       // B*N (logits scratch -> attn)

    // Workspace: partial ctx (8 MB) + per-chunk m/s.
    float* pctx = (float*)d_ws;
    float* pm   = pctx + (size_t)B * CHUNKS * H;
    float* ps   = pm + (size_t)B * CHUNKS;

    dim3 g1(CHUNKS, B);
    flash_partial<<<g1, TPB, 0, stream>>>(outputs, last_h, attn, pctx, pm, ps);
    combine<<<B, 256, 0, stream>>>(pctx, pm, ps, ctx, attn);
}